// DroneGNN_45174466019781
// MI455X (gfx1250) — compile-verified
//
#include <hip/hip_runtime.h>

typedef __attribute__((ext_vector_type(2))) float v2f;
typedef __attribute__((ext_vector_type(8))) float v8f;

#define WAVES 4
#define W1S 65   // padded row stride for sW1 (8x64 zero-padded W1)
#define W2S 65   // padded row stride for sW2 (64x64)
#define H1S 66   // padded row stride for per-wave 16x64 intermediate

__device__ __forceinline__ v8f wmma_f32_k4(v2f a, v2f b, v8f c) {
  // D = A(16x4) * B(4x16) + C(16x16), all f32
  return __builtin_amdgcn_wmma_f32_16x16x4_f32(false, a, false, b, (short)0, c,
                                               false, false);
}

__global__ void zero_f32(float* __restrict__ p, long n) {
  long i = (long)blockIdx.x * blockDim.x + threadIdx.x;
  long stride = (long)gridDim.x * blockDim.x;
  for (; i < n; i += stride) p[i] = 0.f;
}

__global__ void gather_p(const float* __restrict__ h, int stride,
                         float* __restrict__ p, int n) {
  int i = blockIdx.x * blockDim.x + threadIdx.x;
  if (i < n) {
    const float* r = h + (long)i * stride;
    p[3 * i + 0] = r[0];
    p[3 * i + 1] = r[1];
    p[3 * i + 2] = r[14];
  }
}

// Each wave processes 16 edges at a time:
//   stage1: h1 = relu(m(16x6,pad8) @ W1(8x64) + b1)   -> 8 WMMAs (C seeded b1)
//   stage2: o  = h1(16x64) @ W2(64x64) + b2           -> 64 WMMAs (C seeded b2)
//   scatter: atomicAdd into out[dst]
__global__ __launch_bounds__(WAVES * 32) void edge_mlp(
    const float* __restrict__ p, const long long* __restrict__ ei,
    const float* __restrict__ W1, const float* __restrict__ b1,
    const float* __restrict__ W2, const float* __restrict__ b2,
    float* __restrict__ out, int E0, int ET, int nGroups) {
  __shared__ float sW1[8 * W1S];
  __shared__ float sW2[64 * W2S];
  __shared__ float sb1[64];
  __shared__ float sb2[64];
  __shared__ float sh1[WAVES][16 * H1S];
  __shared__ int sdst[WAVES][16];

  const int tid = threadIdx.x;
  for (int i = tid; i < 8 * 64; i += WAVES * 32) {
    int k = i >> 6, n = i & 63;
    sW1[k * W1S + n] = (k < 6) ? W1[k * 64 + n] : 0.f;  // zero-pad K 6->8
  }
  for (int i = tid; i < 64 * 64; i += WAVES * 32) {
    int k = i >> 6, n = i & 63;
    sW2[k * W2S + n] = W2[i];
  }
  if (tid < 64) {
    sb1[tid] = b1[tid];
    sb2[tid] = b2[tid];
  }
  __syncthreads();

  const int wave = tid >> 5;
  const int lane = tid & 31;
  const int l15 = lane & 15;
  const int hi = lane >> 4;  // which half-wave (K-split for A, M-split for C/D)
  const int gStride = gridDim.x * WAVES;
  float* myh1 = &sh1[wave][0];

  for (int g = blockIdx.x * WAVES + wave; g < nGroups; g += gStride) {
    int e = g * 16 + l15;
    bool valid = e < ET;
    int ec = valid ? e : (ET - 1);
    int si, di;
    if (ec < E0) {  // real edge
      si = (int)ei[ec];
      di = (int)ei[(long)E0 + ec];
    } else {  // virtual self-loop edge
      si = ec - E0;
      di = si;
    }
    if (hi == 0) sdst[wave][l15] = valid ? di : -1;

    // prefetch next group's index cachelines (the one HBM-streamed input)
    {
      int gn = g + gStride;
      if (gn * 16 < E0) {
        __builtin_prefetch(&ei[gn * 16 + l15], 0, 1);
        __builtin_prefetch(&ei[(long)E0 + gn * 16 + l15], 0, 1);
      }
    }

    float pi0 = p[di * 3 + 0], pi1 = p[di * 3 + 1], pi2 = p[di * 3 + 2];
    float pj0 = p[si * 3 + 0], pj1 = p[si * 3 + 1], pj2 = p[si * 3 + 2];
    float m0 = pj0 - pi0, m1 = pj1 - pi1, m2 = pj2 - pi2;
    float m3 = pi0, m4 = pi1, m5 = pi2;  // K=6,7 are zero pad

    // ---- stage 1: acc = m @ W1 + b1 (bias via C operand) ----
    v8f acc[4];
#pragma unroll
    for (int nt = 0; nt < 4; ++nt) {
      float bb = sb1[nt * 16 + l15];
#pragma unroll
      for (int r = 0; r < 8; ++r) acc[nt][r] = bb;
    }
    {
      // A layout 16x4 f32: lanes 0-15 hold K={0,1}, lanes 16-31 hold K={2,3}
      v2f A0;
      A0.x = hi ? m2 : m0;
      A0.y = hi ? m3 : m1;
      v2f A1;
      A1.x = hi ? 0.f : m4;
      A1.y = hi ? 0.f : m5;
#pragma unroll
      for (int nt = 0; nt < 4; ++nt) {
        int n = nt * 16 + l15;
        v2f B0;
        B0.x = sW1[(0 + 2 * hi) * W1S + n];
        B0.y = sW1[(1 + 2 * hi) * W1S + n];
        acc[nt] = wmma_f32_k4(A0, B0, acc[nt]);
        v2f B1;
        B1.x = sW1[(4 + 2 * hi) * W1S + n];
        B1.y = sW1[(5 + 2 * hi) * W1S + n];
        acc[nt] = wmma_f32_k4(A1, B1, acc[nt]);
      }
    }

    // relu, spill D-layout tile to LDS (row = r + 8*hi, col = nt*16+l15)
#pragma unroll
    for (int nt = 0; nt < 4; ++nt) {
      int n = nt * 16 + l15;
#pragma unroll
      for (int r = 0; r < 8; ++r) {
        float v = acc[nt][r];
        v = v > 0.f ? v : 0.f;
        myh1[(r + 8 * hi) * H1S + n] = v;
      }
    }
    asm volatile("s_wait_dscnt 0" ::: "memory");

    // ---- stage 2: o = h1 @ W2 + b2 (bias via C operand) ----
    v8f o[4];
#pragma unroll
    for (int nt = 0; nt < 4; ++nt) {
      float bb = sb2[nt * 16 + l15];
#pragma unroll
      for (int r = 0; r < 8; ++r) o[nt][r] = bb;
    }
#pragma unroll
    for (int kt = 0; kt < 16; ++kt) {
      int krow = 4 * kt + 2 * hi;
      v2f A;  // reload in A layout: lane M=l15, K pair split by half-wave
      A.x = myh1[l15 * H1S + krow];
      A.y = myh1[l15 * H1S + krow + 1];
#pragma unroll
      for (int nt = 0; nt < 4; ++nt) {
        int n = nt * 16 + l15;
        v2f B;
        B.x = sW2[krow * W2S + n];
        B.y = sW2[(krow + 1) * W2S + n];
        o[nt] = wmma_f32_k4(A, B, o[nt]);
      }
    }
    asm volatile("s_wait_dscnt 0" ::: "memory");

    // ---- scatter-add (tail rows masked here; EXEC all-ones through WMMA) ----
#pragma unroll
    for (int r = 0; r < 8; ++r) {
      int row = r + 8 * hi;
      int d = sdst[wave][row];
      if (d >= 0) {
        float* po = out + (long)d * 64 + l15;
#pragma unroll
        for (int nt = 0; nt < 4; ++nt) atomicAdd(po + nt * 16, o[nt][r]);
      }
    }
  }
}

__global__ void bn_reduce(const float* __restrict__ h, int n,
                          float* __restrict__ stats) {
  int col = threadIdx.x & 63;
  int rpb = blockDim.x >> 6;
  int row = blockIdx.x * rpb + (threadIdx.x >> 6);
  int rstride = gridDim.x * rpb;
  float s = 0.f, s2 = 0.f;
  for (int r = row; r < n; r += rstride) {
    float v = h[(long)r * 64 + col];
    s += v;
    s2 += v * v;
  }
  atomicAdd(&stats[col], s);
  atomicAdd(&stats[64 + col], s2);
}

__global__ void bn_apply(float* __restrict__ h, long total,
                         const float* __restrict__ stats, float invN,
                         const float* __restrict__ g,
                         const float* __restrict__ b) {
  long i = (long)blockIdx.x * blockDim.x + threadIdx.x;
  if (i < total) {
    int col = (int)(i & 63);
    float mu = stats[col] * invN;
    float var = stats[64 + col] * invN - mu * mu;
    float v = (h[i] - mu) * rsqrtf(var + 1e-5f) * g[col] + b[col];
    h[i] = v > 0.f ? v : 0.f;
  }
}

extern "C" void kernel_launch(void* const* d_in, const int* in_sizes, int n_in,
                              void* d_out, int out_size, void* d_ws,
                              size_t ws_size, hipStream_t stream) {
  const float* x = (const float*)d_in[0];
  const long long* ei = (const long long*)d_in[1];
  const float* W1 = (const float*)d_in[2];     // 6 x 6 x 64
  const float* b1 = (const float*)d_in[3];     // 6 x 64
  const float* W2 = (const float*)d_in[4];     // 6 x 64 x 64
  const float* b2 = (const float*)d_in[5];     // 6 x 64
  const float* gamma = (const float*)d_in[6];  // 5 x 64
  const float* beta = (const float*)d_in[7];   // 5 x 64
  float* out = (float*)d_out;                  // N x 64

  const int N = in_sizes[0] / 16;
  const int E0 = in_sizes[1] / 2;
  const int ET = E0 + N;  // edges + self-loops
  const int nGroups = (ET + 15) / 16;

  float* p = (float*)d_ws;                 // N x 3
  float* bufA = p + ((3 * N + 63) & ~63);  // N x 64
  float* bufB = bufA + (long)N * 64;       // N x 64
  float* stats = bufB + (long)N * 64;      // 128

  const float invN = 1.0f / (float)N;

  const float* hcur = x;
  int stride = 16;
  for (int l = 0; l < 6; ++l) {
    float* hnext = (l == 5) ? out : ((l & 1) ? bufB : bufA);
    gather_p<<<(N + 255) / 256, 256, 0, stream>>>(hcur, stride, p, N);
    zero_f32<<<512, 256, 0, stream>>>(hnext, (long)N * 64);
    zero_f32<<<1, 128, 0, stream>>>(stats, 128);
    edge_mlp<<<1280, WAVES * 32, 0, stream>>>(
        p, ei, W1 + l * 6 * 64, b1 + l * 64, W2 + l * 64 * 64, b2 + l * 64,
        hnext, E0, ET, nGroups);
    if (l < 5) {
      bn_reduce<<<768, 256, 0, stream>>>(hnext, N, stats);
      bn_apply<<<(int)(((long)N * 64 + 255) / 256), 256, 0, stream>>>(
          hnext, (long)N * 64, stats, invN, gamma + l * 64, beta + l * 64);
    }
    hcur = hnext;
    stride = 64;
  }
}